// Network_Nasp_10496900072255
// MI455X (gfx1250) — compile-verified
//
#include <hip/hip_runtime.h>
#include <hip/hip_bf16.h>

#define IN_DIM 128
#define HID    64
#define CNUM   8
#define PLIN   3

typedef float v2f __attribute__((ext_vector_type(2)));
typedef float v4f __attribute__((ext_vector_type(4)));
typedef float v8f __attribute__((ext_vector_type(8)));

// ---------------------------------------------------------------------------
// 1) W_eff[k] = sum_p alphas[k,p] * W_ops[k,p];  b_eff[k] = sum_p a*b_ops;
//    alpha4[k] = alphas[k,3]
// ---------------------------------------------------------------------------
__global__ void k_build_weff(const float* __restrict__ alphas,
                             const float* __restrict__ W_ops,
                             const float* __restrict__ b_ops,
                             float* __restrict__ W_eff,
                             float* __restrict__ b_eff,
                             float* __restrict__ alpha4)
{
    const int k = blockIdx.x;
    const float a0 = alphas[k * 4 + 0];
    const float a1 = alphas[k * 4 + 1];
    const float a2 = alphas[k * 4 + 2];
    const float* w = W_ops + (size_t)k * PLIN * HID * HID;
    float* we = W_eff + (size_t)k * HID * HID;
    for (int i = threadIdx.x; i < HID * HID; i += blockDim.x)
        we[i] = a0 * w[i] + a1 * w[HID * HID + i] + a2 * w[2 * HID * HID + i];
    if (threadIdx.x < HID) {
        const float* b = b_ops + (size_t)k * PLIN * HID;
        b_eff[k * HID + threadIdx.x] = a0 * b[threadIdx.x]
                                     + a1 * b[HID + threadIdx.x]
                                     + a2 * b[2 * HID + threadIdx.x];
    }
    if (threadIdx.x == 0) alpha4[k] = alphas[k * 4 + PLIN];
}

// ---------------------------------------------------------------------------
// 2) M[k] = W_pre(128x64) @ W_eff[k](64x64), stored directly in WMMA
//    B-fragment order:  Bfrag[k][ks][nt][lane]{2}  so each lane's (K,K+1)
//    pair is contiguous (one b64) and a (ks,nt) fragment is one 256B wave
//    fetch.  Also bb[k] = b_pre @ W_eff[k] + b_eff[k].
//
//    Element M[d][h]:  ks = d>>2, r = d&3, nt = h>>4,
//                      lane = (r>>1)*16 + (h&15), comp = r&1.
// ---------------------------------------------------------------------------
__global__ void k_build_m(const float* __restrict__ W_pre,
                          const float* __restrict__ b_pre,
                          const float* __restrict__ W_eff,
                          const float* __restrict__ b_eff,
                          float* __restrict__ Bfrag,
                          float* __restrict__ bb)
{
    const int k = blockIdx.x;
    const float* we = W_eff + (size_t)k * HID * HID;
    for (int i = threadIdx.x; i < IN_DIM * HID; i += blockDim.x) {
        const int d = i / HID, h = i % HID;
        float s = 0.f;
        for (int e = 0; e < HID; ++e) s += W_pre[d * HID + e] * we[e * HID + h];
        const int ks   = d >> 2;
        const int r    = d & 3;
        const int nt   = h >> 4;
        const int lane = ((r >> 1) << 4) + (h & 15);
        const int comp = r & 1;
        const size_t idx = (((((size_t)k * (IN_DIM / 4) + ks) * 4 + nt) * 32) + lane) * 2 + comp;
        Bfrag[idx] = s;
    }
    if (threadIdx.x < HID) {
        const int h = threadIdx.x;
        float s = b_eff[k * HID + h];
        for (int e = 0; e < HID; ++e) s += b_pre[e] * we[e * HID + h];
        bb[k * HID + h] = s;
    }
}

// ---------------------------------------------------------------------------
// 3) bucket bookkeeping
// ---------------------------------------------------------------------------
__global__ void k_init(int* __restrict__ counts)
{
    if (threadIdx.x < 16) counts[threadIdx.x] = 0;
}

__global__ void k_hist(const int* __restrict__ cids, int N, int* __restrict__ counts)
{
    const int n = blockIdx.x * blockDim.x + threadIdx.x;
    if (n >= N) return;
    int c = cids[n];
    if (c < 0 || c > CNUM) c = CNUM;
    atomicAdd(&counts[c], 1);
}

__global__ void k_scan(const int* __restrict__ counts,
                       int* __restrict__ bucketStart,
                       int* __restrict__ cursor,
                       int* __restrict__ tileStart)
{
    if (threadIdx.x == 0) {
        int off = 0, toff = 0;
        for (int k = 0; k <= CNUM; ++k) {
            bucketStart[k] = off;
            cursor[k]      = off;
            tileStart[k]   = toff;
            off  += counts[k];
            toff += (counts[k] + 15) >> 4;
        }
        tileStart[CNUM + 1] = toff;
    }
}

__global__ void k_scatter(const int* __restrict__ cids, int N,
                          int* __restrict__ cursor, int* __restrict__ idx_sorted)
{
    const int n = blockIdx.x * blockDim.x + threadIdx.x;
    if (n >= N) return;
    int c = cids[n];
    if (c < 0 || c > CNUM) c = CNUM;
    const int pos = atomicAdd(&cursor[c], 1);
    idx_sorted[pos] = n;
}

// ---------------------------------------------------------------------------
// 4) Main: one wave32 per 16-node tile of a single cluster.
//    out[16x64] = x_tile[16x128] @ M[k][128x64]  via V_WMMA_F32_16X16X4_F32
// ---------------------------------------------------------------------------
__global__ void __launch_bounds__(32)
k_main(const float* __restrict__ x,
       const float* __restrict__ oneh,
       const float* __restrict__ Bfrag,
       const float* __restrict__ bb,
       const float* __restrict__ alpha4,
       const int* __restrict__ bucketStart,
       const int* __restrict__ counts,
       const int* __restrict__ tileStart,
       const int* __restrict__ idx_sorted,
       float* __restrict__ out)
{
    __shared__ float lds_x[16 * IN_DIM];   // 8 KB staged A-tile

    const int t = blockIdx.x;
    if (t >= tileStart[CNUM + 1]) return;

    int k = 0;
    while (k < CNUM && t >= tileStart[k + 1]) ++k;   // uniform bucket search

    const int tin   = t - tileStart[k];
    const int begin = bucketStart[k] + tin * 16;
    int nrows = counts[k] - tin * 16;
    if (nrows > 16) nrows = 16;

    const int lane = threadIdx.x;

    if (k == CNUM) {
        // sentinel bucket: these nodes are in no cluster -> output 0
        for (int r = 0; r < nrows; ++r) {
            const int node = idx_sorted[begin + r];
            v2f z = {0.f, 0.f};
            *(v2f*)(out + (size_t)node * HID + lane * 2) = z;
        }
        return;
    }

    // ---- stage gathered 16x128 x-tile into LDS (coalesced b128 loads) ----
    for (int r = 0; r < 16; ++r) {
        const int rr = idx_sorted[begin + (r < nrows ? r : 0)];
        v4f vx = *(const v4f*)(x + (size_t)rr * IN_DIM + lane * 4);
        *(v4f*)(&lds_x[r * IN_DIM + lane * 4]) = vx;
    }
    __syncthreads();

    // ---- WMMA fp32 GEMM: 32 K-steps (K=4 each) x 4 N-tiles ----
    // B fragments pre-swizzled: one contiguous v2f per lane per (ks,nt).
    const float* Bk   = Bfrag + (size_t)k * (IN_DIM / 4) * 4 * 32 * 2;
    const int    mrow = lane & 15;          // A row / B,C col within half-wave
    const int    koff = (lane >> 4) * 2;    // hi half-wave carries K+2,K+3

    v8f acc0 = {0,0,0,0,0,0,0,0};
    v8f acc1 = {0,0,0,0,0,0,0,0};
    v8f acc2 = {0,0,0,0,0,0,0,0};
    v8f acc3 = {0,0,0,0,0,0,0,0};

    #pragma unroll
    for (int ks = 0; ks < IN_DIM / 4; ++ks) {
        v2f a = *(const v2f*)(&lds_x[mrow * IN_DIM + ks * 4 + koff]);
        const float* bp = Bk + (size_t)ks * 256 + lane * 2;

        v2f b0 = *(const v2f*)(bp);
        v2f b1 = *(const v2f*)(bp + 64);
        v2f b2 = *(const v2f*)(bp + 128);
        v2f b3 = *(const v2f*)(bp + 192);

        acc0 = __builtin_amdgcn_wmma_f32_16x16x4_f32(false, a, false, b0,
                                                     (short)0, acc0, false, false);
        acc1 = __builtin_amdgcn_wmma_f32_16x16x4_f32(false, a, false, b1,
                                                     (short)0, acc1, false, false);
        acc2 = __builtin_amdgcn_wmma_f32_16x16x4_f32(false, a, false, b2,
                                                     (short)0, acc2, false, false);
        acc3 = __builtin_amdgcn_wmma_f32_16x16x4_f32(false, a, false, b3,
                                                     (short)0, acc3, false, false);
    }

    // ---- epilogue: + bias + alpha4 * one_hot_h, scatter rows ----
    const float a4 = alpha4[k];
    const float* bbk = bb + k * HID;
    const int mbase = (lane >> 4) * 8;      // hi half-wave holds M = j+8

    v8f accs[4] = {acc0, acc1, acc2, acc3};
    #pragma unroll
    for (int nt = 0; nt < 4; ++nt) {
        const int col = nt * 16 + mrow;
        const float bv = bbk[col];
        #pragma unroll
        for (int j = 0; j < 8; ++j) {
            const int m = j + mbase;
            if (m < nrows) {
                const int node = idx_sorted[begin + m];
                out[(size_t)node * HID + col] =
                    accs[nt][j] + bv + a4 * oneh[(size_t)node * HID + col];
            }
        }
    }
}

// ---------------------------------------------------------------------------
extern "C" void kernel_launch(void* const* d_in, const int* in_sizes, int n_in,
                              void* d_out, int out_size, void* d_ws, size_t ws_size,
                              hipStream_t stream)
{
    const float* x      = (const float*)d_in[0];
    const float* oneh   = (const float*)d_in[1];
    const float* alphas = (const float*)d_in[2];
    const float* W_pre  = (const float*)d_in[3];
    const float* b_pre  = (const float*)d_in[4];
    const float* W_ops  = (const float*)d_in[5];
    const float* b_ops  = (const float*)d_in[6];
    const int*   cids   = (const int*)d_in[7];
    float* out = (float*)d_out;

    const int N = in_sizes[0] / IN_DIM;

    // workspace layout (floats, then ints)
    float* wsf    = (float*)d_ws;
    float* W_eff  = wsf;                          // 8*64*64      = 32768 floats
    float* b_eff  = wsf + 32768;                  // 8*64         =   512
    float* alpha4 = wsf + 33280;                  // 8 (pad 16)
    float* bbv    = wsf + 33296;                  // 8*64         =   512
    float* Bfrag  = wsf + 33808;                  // 8*128*64     = 65536
    int*   wsi        = (int*)(wsf + 33808 + CNUM * IN_DIM * HID);
    int*   counts     = wsi;                      // 9 (16 slots)
    int*   bucketSt   = wsi + 16;                 // 9
    int*   cursor     = wsi + 32;                 // 9
    int*   tileStart  = wsi + 48;                 // 10
    int*   idx_sorted = wsi + 64;                 // N

    const int tpb  = 256;
    const int nblk = (N + tpb - 1) / tpb;

    k_build_weff<<<CNUM, 256, 0, stream>>>(alphas, W_ops, b_ops, W_eff, b_eff, alpha4);
    k_build_m   <<<CNUM, 256, 0, stream>>>(W_pre, b_pre, W_eff, b_eff, Bfrag, bbv);
    k_init      <<<1, 64, 0, stream>>>(counts);
    k_hist      <<<nblk, tpb, 0, stream>>>(cids, N, counts);
    k_scan      <<<1, 32, 0, stream>>>(counts, bucketSt, cursor, tileStart);
    k_scatter   <<<nblk, tpb, 0, stream>>>(cids, N, cursor, idx_sorted);

    const int maxTiles = (N + 15) / 16 + CNUM + 1;
    k_main<<<maxTiles, 32, 0, stream>>>(x, oneh, Bfrag, bbv, alpha4,
                                        bucketSt, counts, tileStart, idx_sorted, out);
}